// GenshinAssistant_16423954940425
// MI455X (gfx1250) — compile-verified
//
#include <hip/hip_runtime.h>
#include <hip/hip_bf16.h>
#include <cstddef>

// ---------------------------------------------------------------------------
// Model dimensions (fixed by the reference)
// ---------------------------------------------------------------------------
#define BB 32     // batch
#define SS 256    // src length
#define TT 64     // tgt length
#define TD 63     // decoder steps (T-1)
#define EE 128    // embedding dim
#define HH 256    // encoder per-direction hidden
#define H2 512    // decoder hidden (2*HH)
#define VV 32000  // vocab

typedef __attribute__((ext_vector_type(2))) float v2f;
typedef __attribute__((ext_vector_type(4))) float v4f;
typedef __attribute__((ext_vector_type(8))) float v8f;

// CDNA5 fp32 WMMA: D(16x16,f32) = A(16x4,f32) x B(4x16,f32) + C
__device__ __forceinline__ v8f wmma_f32(v2f a, v2f b, v8f c) {
  return __builtin_amdgcn_wmma_f32_16x16x4_f32(false, a, false, b, (short)0, c,
                                               false, false);
}

// 8 WMMAs covering a K-group of 8 (permuted-K mapping, see analysis):
// each lane holds a float4 of A at [k+4*half .. k+4*half+3] and float4s of
// 4 B tiles at the same K offsets. Micro-step lo consumes .xy, hi consumes .zw
// with identical mappings on A and B, so the contraction is exact.
__device__ __forceinline__ void wmma_group8(const v4f a, const v4f* b,
                                            v8f* acc) {
  v2f alo, ahi;
  alo.x = a.x;
  alo.y = a.y;
  ahi.x = a.z;
  ahi.y = a.w;
#pragma unroll
  for (int j = 0; j < 4; ++j) {
    v2f t;
    t.x = b[j].x;
    t.y = b[j].y;
    acc[j] = wmma_f32(alo, t, acc[j]);
  }
#pragma unroll
  for (int j = 0; j < 4; ++j) {
    v2f t;
    t.x = b[j].z;
    t.y = b[j].w;
    acc[j] = wmma_f32(ahi, t, acc[j]);
  }
}

__device__ __forceinline__ float sigm(float x) {
  return 1.0f / (1.0f + __expf(-x));
}

// ---------------------------------------------------------------------------
// Generic fp32 WMMA GEMM:  C[M,N] (+)= A[M,K] * W[N,K]^T (+bias) (ReLU)
// Each wave computes a 16x64 tile (1 M-tile x 4 N-tiles, A fragment reused).
// Requirements: M % 16 == 0, N % 64 == 0, K % 8 == 0, 16B-aligned rows.
// grid=(N/64, ceil(M/64)), block=128 (4 waves). Software-pipelined b128 loads.
// ---------------------------------------------------------------------------
__global__ __launch_bounds__(128) void gemm_f32(
    const float* __restrict__ A, int lda, const float* __restrict__ W, int ldw,
    const float* __restrict__ bias, float* __restrict__ C, int ldc, int M,
    int N, int K, int accum, int relu) {
  const int wave = threadIdx.x >> 5;
  const int lane = threadIdx.x & 31;
  const int half = lane >> 4;
  const int l = lane & 15;
  const int m0 = (blockIdx.y * 4 + wave) * 16;
  const int n0 = blockIdx.x * 64;
  if (m0 >= M) return;

  v8f acc[4];
#pragma unroll
  for (int j = 0; j < 4; ++j)
#pragma unroll
    for (int r = 0; r < 8; ++r) acc[j][r] = 0.0f;

  const float* arow = A + (size_t)(m0 + l) * lda + 4 * half;
  const float* wrow[4];
#pragma unroll
  for (int j = 0; j < 4; ++j)
    wrow[j] = W + (size_t)(n0 + j * 16 + l) * ldw + 4 * half;

  // prologue
  v4f a_c = *(const v4f*)arow;
  v4f b_c[4];
#pragma unroll
  for (int j = 0; j < 4; ++j) b_c[j] = *(const v4f*)wrow[j];

  for (int k = 8; k <= K - 8; k += 8) {
    // issue next k-group's loads before consuming current fragments
    v4f a_n = *(const v4f*)(arow + k);
    v4f b_n[4];
#pragma unroll
    for (int j = 0; j < 4; ++j) b_n[j] = *(const v4f*)(wrow[j] + k);

    wmma_group8(a_c, b_c, acc);

    a_c = a_n;
#pragma unroll
    for (int j = 0; j < 4; ++j) b_c[j] = b_n[j];
  }
  wmma_group8(a_c, b_c, acc);

#pragma unroll
  for (int j = 0; j < 4; ++j) {
    const int n = n0 + j * 16 + l;
    const float bv = bias ? bias[n] : 0.0f;
#pragma unroll
    for (int r = 0; r < 8; ++r) {
      const int m = m0 + r + 8 * half;
      const size_t off = (size_t)m * ldc + n;
      float v = acc[j][r] + bv;
      if (accum) v += C[off];
      if (relu) v = fmaxf(v, 0.0f);
      C[off] = v;
    }
  }
}

// ---------------------------------------------------------------------------
// Final logits GEMM: out[b, t, :] = hid_all[t*32+b, :] @ out_W2^T + out_b2
// M = TD*BB = 2016, N = VV = 32000, K = HH = 256.
// ---------------------------------------------------------------------------
__global__ __launch_bounds__(128) void logits_gemm(
    const float* __restrict__ A, const float* __restrict__ W,
    const float* __restrict__ bias, float* __restrict__ out) {
  const int wave = threadIdx.x >> 5;
  const int lane = threadIdx.x & 31;
  const int half = lane >> 4;
  const int l = lane & 15;
  const int m0 = (blockIdx.y * 4 + wave) * 16;
  const int n0 = blockIdx.x * 64;
  if (m0 >= TD * BB) return;

  v8f acc[4];
#pragma unroll
  for (int j = 0; j < 4; ++j)
#pragma unroll
    for (int r = 0; r < 8; ++r) acc[j][r] = 0.0f;

  const float* arow = A + (size_t)(m0 + l) * HH + 4 * half;
  const float* wrow[4];
#pragma unroll
  for (int j = 0; j < 4; ++j)
    wrow[j] = W + (size_t)(n0 + j * 16 + l) * HH + 4 * half;

  v4f a_c = *(const v4f*)arow;
  v4f b_c[4];
#pragma unroll
  for (int j = 0; j < 4; ++j) b_c[j] = *(const v4f*)wrow[j];

  for (int k = 8; k <= HH - 8; k += 8) {
    v4f a_n = *(const v4f*)(arow + k);
    v4f b_n[4];
#pragma unroll
    for (int j = 0; j < 4; ++j) b_n[j] = *(const v4f*)(wrow[j] + k);

    wmma_group8(a_c, b_c, acc);

    a_c = a_n;
#pragma unroll
    for (int j = 0; j < 4; ++j) b_c[j] = b_n[j];
  }
  wmma_group8(a_c, b_c, acc);

#pragma unroll
  for (int j = 0; j < 4; ++j) {
    const int n = n0 + j * 16 + l;
    const float bv = bias[n];
#pragma unroll
    for (int r = 0; r < 8; ++r) {
      const int m = m0 + r + 8 * half;  // m = t*32 + b
      const int t = m >> 5;
      const int b = m & 31;
      out[(size_t)b * TD * VV + (size_t)t * VV + n] = acc[j][r] + bv;
    }
  }
}

// ---------------------------------------------------------------------------
// Persistent single-workgroup LSTM recurrence (encoder, H = 256).
// Xpre[(t*32+b), 4*HH] holds x_t @ Wih^T (precomputed).  h lives in LDS
// (read back as b128 ds loads), c in WMMA-layout registers.  16 waves; wave w
// owns h-columns [16w,16w+16) and all 4 gates for them -> wave-local update.
// ---------------------------------------------------------------------------
__global__ __launch_bounds__(512) void enc_lstm(
    const float* __restrict__ Xpre,   // (SS*BB, 4*HH) time-major
    const float* __restrict__ Whh,    // (4*HH, HH)
    const float* __restrict__ bias,   // (4*HH)
    float* __restrict__ Y, int ycol,  // (SS*BB, 2*HH)
    float* __restrict__ hFin, float* __restrict__ cFin, int fincol,  // (BB,H2)
    int reverse) {
  __shared__ float hsh[32][264];  // 32 x 256 h, padded rows (264 % 4 == 0)
  const int wave = threadIdx.x >> 5;
  const int lane = threadIdx.x & 31;
  const int half = lane >> 4;
  const int l = lane & 15;
  const int j0 = wave * 16;

  for (int i = threadIdx.x; i < 32 * 264; i += 512) (&hsh[0][0])[i] = 0.0f;

  float creg[2][8];
  float hreg[2][8];
#pragma unroll
  for (int mt = 0; mt < 2; ++mt)
#pragma unroll
    for (int r = 0; r < 8; ++r) creg[mt][r] = 0.0f;

  float bg[4];
#pragma unroll
  for (int gi = 0; gi < 4; ++gi) bg[gi] = bias[gi * HH + j0 + l];

  const float* wrow[4];
#pragma unroll
  for (int gi = 0; gi < 4; ++gi)
    wrow[gi] = Whh + (size_t)(gi * HH + j0 + l) * HH + 4 * half;

  __syncthreads();

  for (int stepi = 0; stepi < SS; ++stepi) {
    const int t = reverse ? (SS - 1 - stepi) : stepi;
#pragma unroll
    for (int mt = 0; mt < 2; ++mt) {
      const int m0 = mt * 16;
      v8f acc[4];
#pragma unroll
      for (int gi = 0; gi < 4; ++gi)
#pragma unroll
        for (int r = 0; r < 8; ++r) acc[gi][r] = 0.0f;

      const float* hrow = &hsh[m0 + l][4 * half];

      v4f a_c = *(const v4f*)hrow;  // ds_load_b128
      v4f b_c[4];
#pragma unroll
      for (int gi = 0; gi < 4; ++gi) b_c[gi] = *(const v4f*)wrow[gi];

      for (int k = 8; k <= HH - 8; k += 8) {
        v4f a_n = *(const v4f*)(hrow + k);
        v4f b_n[4];
#pragma unroll
        for (int gi = 0; gi < 4; ++gi) b_n[gi] = *(const v4f*)(wrow[gi] + k);

        wmma_group8(a_c, b_c, acc);

        a_c = a_n;
#pragma unroll
        for (int gi = 0; gi < 4; ++gi) b_c[gi] = b_n[gi];
      }
      wmma_group8(a_c, b_c, acc);

#pragma unroll
      for (int r = 0; r < 8; ++r) {
        const int m = m0 + r + 8 * half;
        const float* xr = Xpre + ((size_t)t * BB + m) * (4 * HH);
        const float zi = acc[0][r] + xr[0 * HH + j0 + l] + bg[0];
        const float zf = acc[1][r] + xr[1 * HH + j0 + l] + bg[1];
        const float zg = acc[2][r] + xr[2 * HH + j0 + l] + bg[2];
        const float zo = acc[3][r] + xr[3 * HH + j0 + l] + bg[3];
        const float cn = sigm(zf) * creg[mt][r] + sigm(zi) * tanhf(zg);
        creg[mt][r] = cn;
        const float hn = sigm(zo) * tanhf(cn);
        hreg[mt][r] = hn;
        Y[((size_t)t * BB + m) * (2 * HH) + ycol + j0 + l] = hn;
      }
    }
    __syncthreads();  // all waves done reading h
#pragma unroll
    for (int mt = 0; mt < 2; ++mt)
#pragma unroll
      for (int r = 0; r < 8; ++r)
        hsh[mt * 16 + r + 8 * half][j0 + l] = hreg[mt][r];
    __syncthreads();  // new h visible
  }

#pragma unroll
  for (int mt = 0; mt < 2; ++mt)
#pragma unroll
    for (int r = 0; r < 8; ++r) {
      const int m = mt * 16 + r + 8 * half;
      hFin[(size_t)m * H2 + fincol + j0 + l] = hreg[mt][r];
      cFin[(size_t)m * H2 + fincol + j0 + l] = creg[mt][r];
    }
}

// ---------------------------------------------------------------------------
// Embedding gathers (time-major rows: row = t*32 + b), float4 per thread
// ---------------------------------------------------------------------------
__global__ __launch_bounds__(256) void embed_src_k(
    const int* __restrict__ src, const float* __restrict__ emb,
    float* __restrict__ out) {
  const int i = blockIdx.x * 256 + threadIdx.x;  // float4 index
  if (i >= BB * SS * EE / 4) return;
  const int e4 = i & (EE / 4 - 1);
  const int row = i >> 5;  // EE/4 == 32
  const int b = row & 31, s = row >> 5;
  const v4f* erow = (const v4f*)(emb + (size_t)src[b * SS + s] * EE);
  ((v4f*)out)[i] = erow[e4];
}

__global__ __launch_bounds__(256) void embed_tgt_k(
    const int* __restrict__ tgt, const float* __restrict__ emb,
    float* __restrict__ out) {
  const int i = blockIdx.x * 256 + threadIdx.x;  // float4 index
  if (i >= TD * BB * EE / 4) return;
  const int e4 = i & (EE / 4 - 1);
  const int row = i >> 5;
  const int b = row & 31, t = row >> 5;
  const v4f* erow = (const v4f*)(emb + (size_t)tgt[b * TT + t] * EE);
  ((v4f*)out)[i] = erow[e4];
}

// ---------------------------------------------------------------------------
// Attention scores: scores[b,s] = tanh(enc_proj[s*32+b,:] + dec_part[b,:]).W2
//                   + b2.  One wave per (b,s); 8 waves per block.
// ---------------------------------------------------------------------------
__global__ __launch_bounds__(256) void attn_scores_k(
    const float* __restrict__ enc_proj, const float* __restrict__ dec_part,
    const float* __restrict__ W2, const float* __restrict__ b2,
    float* __restrict__ scores) {
  const int wave = threadIdx.x >> 5;
  const int lane = threadIdx.x & 31;
  const int p = blockIdx.x * 8 + wave;  // p = s*32 + b
  const int b = p & 31, s = p >> 5;
  const float* ep = enc_proj + (size_t)p * HH;
  const float* dp = dec_part + (size_t)b * HH;
  float sum = 0.0f;
  for (int k = lane; k < HH; k += 32) sum += tanhf(ep[k] + dp[k]) * W2[k];
#pragma unroll
  for (int off = 16; off > 0; off >>= 1) sum += __shfl_xor(sum, off, 32);
  if (lane == 0) scores[b * SS + s] = sum + b2[0];
}

// ---------------------------------------------------------------------------
// Per-batch softmax over S and context reduction ctx[b,:] = sum_s a[s]*enc_out
// ---------------------------------------------------------------------------
__global__ __launch_bounds__(256) void softmax_ctx_k(
    const float* __restrict__ scores, const float* __restrict__ enc_out,
    float* __restrict__ ctx) {
  __shared__ float sh[SS];
  __shared__ float red[SS];
  const int b = blockIdx.x;
  const int tid = threadIdx.x;
  const float sc = scores[b * SS + tid];
  red[tid] = sc;
  __syncthreads();
  for (int o = 128; o > 0; o >>= 1) {
    if (tid < o) red[tid] = fmaxf(red[tid], red[tid + o]);
    __syncthreads();
  }
  const float mx = red[0];
  __syncthreads();
  const float ex = __expf(sc - mx);
  sh[tid] = ex;
  red[tid] = ex;
  __syncthreads();
  for (int o = 128; o > 0; o >>= 1) {
    if (tid < o) red[tid] += red[tid + o];
    __syncthreads();
  }
  const float inv = 1.0f / red[0];

  float a0 = 0.0f, a1 = 0.0f;
  for (int s = 0; s < SS; ++s) {
    const float a = sh[s] * inv;
    const float* er = enc_out + ((size_t)s * BB + b) * (2 * HH);
    a0 += a * er[tid];
    a1 += a * er[tid + HH];
  }
  ctx[(size_t)b * (2 * HH) + tid] = a0;
  ctx[(size_t)b * (2 * HH) + HH + tid] = a1;
}

// ---------------------------------------------------------------------------
// Decoder LSTM cell activation: z = gates (+ pre); update h,c (H2 = 512)
// ---------------------------------------------------------------------------
__global__ __launch_bounds__(256) void lstm_act_k(
    const float* __restrict__ gates, const float* __restrict__ pre,
    float* __restrict__ h, float* __restrict__ c) {
  const int i = blockIdx.x * 256 + threadIdx.x;
  if (i >= BB * H2) return;
  const int b = i >> 9;
  const int j = i & (H2 - 1);
  const float* z = gates + (size_t)b * 4 * H2;
  float zi = z[j], zf = z[H2 + j], zg = z[2 * H2 + j], zo = z[3 * H2 + j];
  if (pre) {
    const float* p = pre + (size_t)b * 4 * H2;
    zi += p[j];
    zf += p[H2 + j];
    zg += p[2 * H2 + j];
    zo += p[3 * H2 + j];
  }
  const float cn = sigm(zf) * c[i] + sigm(zi) * tanhf(zg);
  c[i] = cn;
  h[i] = sigm(zo) * tanhf(cn);
}

// ---------------------------------------------------------------------------
// Host orchestration
// ---------------------------------------------------------------------------
extern "C" void kernel_launch(void* const* d_in, const int* in_sizes, int n_in,
                              void* d_out, int out_size, void* d_ws,
                              size_t ws_size, hipStream_t stream) {
  const int* src = (const int*)d_in[0];
  const int* tgt = (const int*)d_in[1];
  const float* emb = (const float*)d_in[2];
  const float* eWih0 = (const float*)d_in[3];   // (2, 1024, 128)
  const float* eWhh0 = (const float*)d_in[4];   // (2, 1024, 256)
  const float* eb0 = (const float*)d_in[5];     // (2, 1024)
  const float* eWih1 = (const float*)d_in[6];   // (2, 1024, 512)
  const float* eWhh1 = (const float*)d_in[7];   // (2, 1024, 256)
  const float* eb1 = (const float*)d_in[8];     // (2, 1024)
  const float* dWih0 = (const float*)d_in[9];   // (2048, 640)
  const float* dWhh0 = (const float*)d_in[10];  // (2048, 512)
  const float* db0 = (const float*)d_in[11];    // (2048)
  const float* dWih1 = (const float*)d_in[12];  // (2048, 512)
  const float* dWhh1 = (const float*)d_in[13];  // (2048, 512)
  const float* db1 = (const float*)d_in[14];    // (2048)
  const float* aW1 = (const float*)d_in[15];    // (256, 1024)
  const float* ab1 = (const float*)d_in[16];    // (256)
  const float* aW2 = (const float*)d_in[17];    // (256)
  const float* ab2 = (const float*)d_in[18];    // (1)
  const float* oW1 = (const float*)d_in[19];    // (256, 512)
  const float* ob1 = (const float*)d_in[20];    // (256)
  const float* oW2 = (const float*)d_in[21];    // (32000, 256)
  const float* ob2 = (const float*)d_in[22];    // (32000)
  float* out = (float*)d_out;
  (void)in_sizes;
  (void)n_in;
  (void)out_size;
  (void)ws_size;

  // Workspace bump allocator (floats, 64-element aligned)
  float* w = (float*)d_ws;
  size_t off = 0;
  auto alloc = [&](size_t n) {
    float* p = w + off;
    off += (n + 63) & ~(size_t)63;
    return p;
  };
  float* src_e = alloc((size_t)BB * SS * EE);
  float* Xa = alloc((size_t)BB * SS * 4 * HH);  // layer0 fwd, reused layer1 fwd
  float* Xb = alloc((size_t)BB * SS * 4 * HH);  // layer0 bwd, reused layer1 bwd
  float* y0 = alloc((size_t)BB * SS * 2 * HH);
  float* enc_out = alloc((size_t)BB * SS * 2 * HH);
  float* enc_proj = alloc((size_t)BB * SS * HH);
  float* tgt_e = alloc((size_t)TD * BB * EE);
  float* tgt_pre = alloc((size_t)TD * BB * 4 * H2);
  float* h0 = alloc((size_t)BB * H2);
  float* c0 = alloc((size_t)BB * H2);
  float* h1 = alloc((size_t)BB * H2);
  float* c1 = alloc((size_t)BB * H2);
  float* dec_part = alloc((size_t)BB * HH);
  float* scores = alloc((size_t)BB * SS);
  float* ctx = alloc((size_t)BB * H2);
  float* gates0 = alloc((size_t)BB * 4 * H2);
  float* gates1 = alloc((size_t)BB * 4 * H2);
  float* hid_all = alloc((size_t)TD * BB * HH);

  const dim3 blk128(128), blk256(256), blk512(512);

  // --- embeddings -------------------------------------------------------
  embed_src_k<<<(BB * SS * EE / 4 + 255) / 256, blk256, 0, stream>>>(src, emb,
                                                                     src_e);
  embed_tgt_k<<<(TD * BB * EE / 4 + 255) / 256, blk256, 0, stream>>>(tgt, emb,
                                                                     tgt_e);

  // --- encoder layer 0: input projections (big parallel WMMA GEMMs) -----
  gemm_f32<<<dim3(4 * HH / 64, BB * SS / 64), blk128, 0, stream>>>(
      src_e, EE, eWih0, EE, nullptr, Xa, 4 * HH, BB * SS, 4 * HH, EE, 0, 0);
  gemm_f32<<<dim3(4 * HH / 64, BB * SS / 64), blk128, 0, stream>>>(
      src_e, EE, eWih0 + (size_t)4 * HH * EE, EE, nullptr, Xb, 4 * HH, BB * SS,
      4 * HH, EE, 0, 0);
  // decoder tgt-embedding projection (first EE cols of dWih0), done up front
  gemm_f32<<<dim3(4 * H2 / 64, (TD * BB + 63) / 64), blk128, 0, stream>>>(
      tgt_e, EE, dWih0, EE + H2, nullptr, tgt_pre, 4 * H2, TD * BB, 4 * H2, EE,
      0, 0);

  // --- encoder layer 0 recurrences (persistent, single workgroup each) --
  enc_lstm<<<1, blk512, 0, stream>>>(Xa, eWhh0, eb0, y0, 0, h0, c0, 0, 0);
  enc_lstm<<<1, blk512, 0, stream>>>(Xb, eWhh0 + (size_t)4 * HH * HH,
                                     eb0 + 4 * HH, y0, HH, h0, c0, HH, 1);

  // --- encoder layer 1 --------------------------------------------------
  gemm_f32<<<dim3(4 * HH / 64, BB * SS / 64), blk128, 0, stream>>>(
      y0, 2 * HH, eWih1, 2 * HH, nullptr, Xa, 4 * HH, BB * SS, 4 * HH, 2 * HH,
      0, 0);
  gemm_f32<<<dim3(4 * HH / 64, BB * SS / 64), blk128, 0, stream>>>(
      y0, 2 * HH, eWih1 + (size_t)4 * HH * 2 * HH, 2 * HH, nullptr, Xb, 4 * HH,
      BB * SS, 4 * HH, 2 * HH, 0, 0);
  enc_lstm<<<1, blk512, 0, stream>>>(Xa, eWhh1, eb1, enc_out, 0, h1, c1, 0, 0);
  enc_lstm<<<1, blk512, 0, stream>>>(Xb, eWhh1 + (size_t)4 * HH * HH,
                                     eb1 + 4 * HH, enc_out, HH, h1, c1, HH, 1);

  // --- attention precompute: enc_proj = enc_out @ W1_enc^T + ab1 --------
  gemm_f32<<<dim3(HH / 64, BB * SS / 64), blk128, 0, stream>>>(
      enc_out, 2 * HH, aW1 + H2, 4 * HH, ab1, enc_proj, HH, BB * SS, HH, H2, 0,
      0);

  // --- decoder loop -----------------------------------------------------
  for (int t = 0; t < TD; ++t) {
    // dec_part = h1 @ W1_dec^T
    gemm_f32<<<dim3(HH / 64, 1), blk128, 0, stream>>>(
        h1, H2, aW1, 4 * HH, nullptr, dec_part, HH, BB, HH, H2, 0, 0);
    attn_scores_k<<<BB * SS / 8, blk256, 0, stream>>>(enc_proj, dec_part, aW2,
                                                      ab2, scores);
    softmax_ctx_k<<<BB, blk256, 0, stream>>>(scores, enc_out, ctx);

    // layer-0 cell: gates0 = ctx @ Wih0[:,128:]^T + b0 + h0 @ Whh0^T
    gemm_f32<<<dim3(4 * H2 / 64, 1), blk128, 0, stream>>>(
        ctx, H2, dWih0 + EE, EE + H2, db0, gates0, 4 * H2, BB, 4 * H2, H2, 0,
        0);
    gemm_f32<<<dim3(4 * H2 / 64, 1), blk128, 0, stream>>>(
        h0, H2, dWhh0, H2, nullptr, gates0, 4 * H2, BB, 4 * H2, H2, 1, 0);
    lstm_act_k<<<BB * H2 / 256, blk256, 0, stream>>>(
        gates0, tgt_pre + (size_t)t * BB * 4 * H2, h0, c0);

    // layer-1 cell: gates1 = h0 @ Wih1^T + b1 + h1 @ Whh1^T
    gemm_f32<<<dim3(4 * H2 / 64, 1), blk128, 0, stream>>>(
        h0, H2, dWih1, H2, db1, gates1, 4 * H2, BB, 4 * H2, H2, 0, 0);
    gemm_f32<<<dim3(4 * H2 / 64, 1), blk128, 0, stream>>>(
        h1, H2, dWhh1, H2, nullptr, gates1, 4 * H2, BB, 4 * H2, H2, 1, 0);
    lstm_act_k<<<BB * H2 / 256, blk256, 0, stream>>>(gates1, nullptr, h1, c1);

    // hid_t = relu(h1 @ out_W1^T + out_b1), deferred logits
    gemm_f32<<<dim3(HH / 64, 1), blk128, 0, stream>>>(
        h1, H2, oW1, H2, ob1, hid_all + (size_t)t * BB * HH, HH, BB, HH, H2, 0,
        1);
  }

  // --- one big deferred logits GEMM (33 GFLOP, device-wide) -------------
  logits_gemm<<<dim3(VV / 64, (TD * BB + 63) / 64), blk128, 0, stream>>>(
      hid_all, oW2, ob2, out);
}